// Net_56152402427955
// MI455X (gfx1250) — compile-verified
//
#include <hip/hip_runtime.h>
#include <math.h>
#include <stdint.h>

// ---------------------------------------------------------------------------
// GAT 2-layer link predictor for MI455X (gfx1250, wave32).
//  * GEMMs: V_WMMA_F32_16X16X4_F32 (full fp32 matrix path — workload is
//    scatter-bound at ~3.5 GB edge traffic, so fp32 WMMA is both fastest
//    and numerically faithful).
//  * A/B panels staged into LDS once per block by the Tensor Data Mover
//    (TENSOR_LOAD_TO_LDS + s_wait_tensorcnt), so the WMMA K-loop runs
//    entirely out of LDS.
// ---------------------------------------------------------------------------

#define NEG_SLOPE 0.2f
#define GAT_EPS   1e-16f

typedef float        v2f __attribute__((ext_vector_type(2)));
typedef float        v8f __attribute__((ext_vector_type(8)));
typedef unsigned int v4u __attribute__((ext_vector_type(4)));
typedef int          v4i __attribute__((ext_vector_type(4)));
typedef int          v8i __attribute__((ext_vector_type(8)));

// ---------------------------------------------------------------------------
// Tensor Data Mover: 2D tile (tile_d1 rows x tile_d0 elems) of a fp32 tensor
// (row stride = stride0 elems) -> contiguous LDS at lds_addr.
// D# bit layout per CDNA5 ISA.
// ---------------------------------------------------------------------------
__device__ __forceinline__ void tdm_load_2d_f32(uint32_t lds_addr, const void* gptr,
                                                uint32_t tensor_d0, uint32_t tensor_d1,
                                                uint32_t tile_d0, uint32_t tile_d1,
                                                uint64_t stride0_elems)
{
    const uint64_t ga = (uint64_t)(uintptr_t)gptr;
    v4u g0;
    g0.x = 1u;                                                // count=1, no gather
    g0.y = lds_addr;                                          // LDS byte address
    g0.z = (uint32_t)ga;                                      // global_addr[31:0]
    g0.w = (uint32_t)((ga >> 32) & 0x01FFFFFFu) | (2u << 30); // addr[56:32] | type=2

    v8i g1;
    g1[0] = (int)(2u << 16);                                  // data_size = 4B
    g1[1] = (int)((tensor_d0 & 0xFFFFu) << 16);               // tensor_dim0[15:0]
    g1[2] = (int)((tensor_d0 >> 16) | ((tensor_d1 & 0xFFFFu) << 16));
    g1[3] = (int)((tensor_d1 >> 16) | (tile_d0 << 16));       // tile_dim0
    g1[4] = (int)(tile_d1 & 0xFFFFu);                         // tile_dim1 (tile_dim2=0)
    g1[5] = (int)(uint32_t)(stride0_elems & 0xFFFFFFFFu);     // dim0_stride lo
    g1[6] = (int)(uint32_t)((stride0_elems >> 32) & 0xFFFFu); // dim0_stride hi
    g1[7] = 0;                                                // dim1_stride unused (2D)

    const v4i z4 = {0, 0, 0, 0};
#if __clang_major__ >= 23
    const v8i z8 = {0, 0, 0, 0, 0, 0, 0, 0};
    __builtin_amdgcn_tensor_load_to_lds(g0, g1, z4, z4, z8, 0);
#else
    __builtin_amdgcn_tensor_load_to_lds(g0, g1, z4, z4, 0);
#endif
}

// ---------------------------------------------------------------------------
// Y[N,Dout] = X[N,K] @ W[K,Dout], fp32 WMMA 16x16x4, panels staged via TDM.
// Block = 4 waves; block tile = 16 rows x 64 cols; K <= 256.
// Fragment layouts per CDNA5 ISA §7.12.2 (wave32):
//   A 16x4 : lane r(0..15) = row; VGPR v holds K = k + 2*half + v (half=lane>>4)
//   B 4x16 : lane r(0..15) = col; VGPR v holds K = k + 2*half + v
//   D 16x16: VGPR g -> row = g + 8*half, col = lane&15
// ---------------------------------------------------------------------------
#define GEMM_WAVES 4
#define KMAX 256
__global__ __launch_bounds__(32 * GEMM_WAVES)
void gemm_wmma_f32(const float* __restrict__ X, const float* __restrict__ W,
                   float* __restrict__ Y, int N, int K, int Dout)
{
    __shared__ float sA[16 * KMAX];               // 16 KB
    __shared__ float sB[KMAX * 16 * GEMM_WAVES];  // 64 KB

    const int lane = threadIdx.x & 31;
    const int wave = threadIdx.x >> 5;
    const int half = lane >> 4;
    const int r    = lane & 15;
    const int m0   = blockIdx.y * 16;
    const int nblk = blockIdx.x * (16 * GEMM_WAVES);
    const int n0   = nblk + wave * 16;

    if (wave == 0) {
        // A panel: 16 rows x K of X  -> sA[row*K + k]
        tdm_load_2d_f32((uint32_t)(uintptr_t)sA, X + (size_t)m0 * K,
                        (uint32_t)K, (uint32_t)N, (uint32_t)K, 16u, (uint64_t)K);
        // B panel: K rows x 64 cols of W -> sB[k*64 + col]
        tdm_load_2d_f32((uint32_t)(uintptr_t)sB, W + nblk,
                        (uint32_t)Dout, (uint32_t)K, 16u * GEMM_WAVES, (uint32_t)K,
                        (uint64_t)Dout);
        __builtin_amdgcn_s_wait_tensorcnt(0);
    }
    __syncthreads();

    const float* __restrict__ aPtr = sA + (size_t)r * K;          // A row for lane
    const float* __restrict__ bPtr = sB + (wave * 16 + r);        // B col for lane

    v8f c = {};
    for (int k = 0; k < K; k += 4) {
        v2f a = *(const v2f*)(aPtr + k + 2 * half);               // ds_load_b64
        v2f b;
        b.x = bPtr[(k + 2 * half    ) * (16 * GEMM_WAVES)];       // ds_load_b32
        b.y = bPtr[(k + 2 * half + 1) * (16 * GEMM_WAVES)];
        c = __builtin_amdgcn_wmma_f32_16x16x4_f32(
                /*neg_a=*/false, a, /*neg_b=*/false, b,
                /*c_mod=*/(short)0, c, /*reuse_a=*/false, /*reuse_b=*/false);
    }
#pragma unroll
    for (int g = 0; g < 8; ++g)
        Y[(size_t)(m0 + g + 8 * half) * Dout + n0 + r] = c[g];
}

// ---------------------------------------------------------------------------
// Per-(node, head) attention scores + softmax-state init. Wave per (n,h).
// ---------------------------------------------------------------------------
__global__ void node_scores(const float* __restrict__ xw,
                            const float* __restrict__ att_src,
                            const float* __restrict__ att_dst,
                            float* __restrict__ a_src, float* __restrict__ a_dst,
                            float* __restrict__ amax, float* __restrict__ denom,
                            int N, int H, int C)
{
    const int gid  = (int)((blockIdx.x * (long long)blockDim.x + threadIdx.x) >> 5);
    const int lane = threadIdx.x & 31;
    if (gid >= N * H) return;
    const int n = gid / H;
    const int h = gid - n * H;
    const float* __restrict__ row = xw + (size_t)n * H * C + (size_t)h * C;
    const float* __restrict__ as  = att_src + h * C;
    const float* __restrict__ ad  = att_dst + h * C;
    float ss = 0.f, sd = 0.f;
    for (int ci = lane; ci < C; ci += 32) {
        const float v = row[ci];
        ss += v * as[ci];
        sd += v * ad[ci];
    }
    for (int off = 16; off; off >>= 1) {
        ss += __shfl_down(ss, off, 32);
        sd += __shfl_down(sd, off, 32);
    }
    if (lane == 0) {
        a_src[gid] = ss;
        a_dst[gid] = sd;
        amax[gid]  = -__builtin_inff();
        denom[gid] = 0.f;
    }
}

__device__ __forceinline__ float leaky(float v) {
    return v > 0.f ? v : NEG_SLOPE * v;
}

__device__ __forceinline__ void atomicMaxF(float* addr, float val) {
    int cur = __float_as_int(*addr);
    while (__int_as_float(cur) < val) {
        const int old = atomicCAS((int*)addr, cur, __float_as_int(val));
        if (old == cur) break;
        cur = old;
    }
}

// Edge enumeration includes PyG self-loops: e in [0, E+N), e>=E -> (e-E, e-E).
__device__ __forceinline__ void edge_sd(const int* __restrict__ ei, int E, int e,
                                        int& s, int& d) {
    if (e < E) { s = ei[e]; d = ei[E + e]; }
    else       { s = e - E; d = e - E; }
}

// Pass 1: segment max of leaky_relu(a_src[src]+a_dst[dst]) over dst.
__global__ void edge_softmax_max(const int* __restrict__ ei, int E, int N,
                                 const float* __restrict__ a_src,
                                 const float* __restrict__ a_dst,
                                 float* __restrict__ amax, int H)
{
    const long long idx   = blockIdx.x * (long long)blockDim.x + threadIdx.x;
    const long long total = (long long)(E + N) * H;
    if (idx >= total) return;
    const int e = (int)(idx / H);
    const int h = (int)(idx - (long long)e * H);
    int s, d;
    edge_sd(ei, E, e, s, d);
    const float al = leaky(a_src[s * H + h] + a_dst[d * H + h]);
    atomicMaxF(&amax[d * H + h], al);
}

// Pass 2: denom[dst] += exp(alpha - amax[dst]).
__global__ void edge_softmax_sum(const int* __restrict__ ei, int E, int N,
                                 const float* __restrict__ a_src,
                                 const float* __restrict__ a_dst,
                                 const float* __restrict__ amax,
                                 float* __restrict__ denom, int H)
{
    const long long idx   = blockIdx.x * (long long)blockDim.x + threadIdx.x;
    const long long total = (long long)(E + N) * H;
    if (idx >= total) return;
    const int e = (int)(idx / H);
    const int h = (int)(idx - (long long)e * H);
    int s, d;
    edge_sd(ei, E, e, s, d);
    const float al = leaky(a_src[s * H + h] + a_dst[d * H + h]);
    atomicAdd(&denom[d * H + h], expf(al - amax[d * H + h]));
}

// Pass 3: out[dst, h, c] += xw[src, h, c] * softmax_weight(edge, h).
// One block (256 threads) per edge; thread t owns feature t (h = t/C).
__global__ void edge_aggregate(const int* __restrict__ ei, int E, int N,
                               const float* __restrict__ xw,
                               const float* __restrict__ a_src,
                               const float* __restrict__ a_dst,
                               const float* __restrict__ amax,
                               const float* __restrict__ denom,
                               float* __restrict__ out, int H, int C)
{
    const int e = blockIdx.x;
    const int t = threadIdx.x;             // blockDim.x == H*C == 256
    const int h = t / C;
    int s, d;
    edge_sd(ei, E, e, s, d);
    const float* __restrict__ srow = xw + (size_t)s * H * C;
    __builtin_prefetch(srow + t, 0, 0);    // global_prefetch_b8: hide gather latency
    const float al = leaky(a_src[s * H + h] + a_dst[d * H + h]);
    const float w  = expf(al - amax[d * H + h]) / (denom[d * H + h] + GAT_EPS);
    atomicAdd(&out[(size_t)d * H * C + t], srow[t] * w);
}

__global__ void zero_f32(float* __restrict__ p, long long n)
{
    const long long i = blockIdx.x * (long long)blockDim.x + threadIdx.x;
    if (i < n) p[i] = 0.f;
}

// io[n,col] = act(io[n,col] + bias[col])
__global__ void bias_act(float* __restrict__ io, const float* __restrict__ bias,
                         long long total, int D, int do_relu)
{
    const long long i = blockIdx.x * (long long)blockDim.x + threadIdx.x;
    if (i >= total) return;
    const int col = (int)(i % D);
    float v = io[i] + bias[col];
    if (do_relu) v = fmaxf(v, 0.f);
    io[i] = v;
}

// logits[i] = <z[u_i], z[v_i]>, one wave per label pair.
__global__ void decode_dot(const float* __restrict__ z, const int* __restrict__ eli,
                           int EL, int D, float* __restrict__ out)
{
    const int gid  = (int)((blockIdx.x * (long long)blockDim.x + threadIdx.x) >> 5);
    const int lane = threadIdx.x & 31;
    if (gid >= EL) return;
    const float* __restrict__ zu = z + (size_t)eli[gid] * D;
    const float* __restrict__ zv = z + (size_t)eli[EL + gid] * D;
    float ssum = 0.f;
    for (int c = lane; c < D; c += 32) ssum += zu[c] * zv[c];
    for (int off = 16; off; off >>= 1) ssum += __shfl_down(ssum, off, 32);
    if (lane == 0) out[gid] = ssum;
}

// ---------------------------------------------------------------------------
extern "C" void kernel_launch(void* const* d_in, const int* in_sizes, int n_in,
                              void* d_out, int out_size, void* d_ws, size_t ws_size,
                              hipStream_t stream)
{
    const float* x   = (const float*)d_in[0];
    const int*   ei  = (const int*)  d_in[1];
    const int*   eli = (const int*)  d_in[2];
    const float* W1  = (const float*)d_in[3];
    const float* as1 = (const float*)d_in[4];
    const float* ad1 = (const float*)d_in[5];
    const float* b1  = (const float*)d_in[6];
    const float* W2  = (const float*)d_in[7];
    const float* as2 = (const float*)d_in[8];
    const float* ad2 = (const float*)d_in[9];
    const float* b2  = (const float*)d_in[10];
    float* logits = (float*)d_out;

    const int N   = in_sizes[0] / 128;   // 50000
    const int E   = in_sizes[1] / 2;     // 800000
    const int EL  = in_sizes[2] / 2;     // 100000
    const int Fin = 128, D = 256, H1 = 4, C1 = 64;
    const int ET  = E + N;               // edges incl. self-loops

    // ---- workspace carve-out (floats): 3 big buffers + 4 score buffers
    float* ws    = (float*)d_ws;
    float* bufA  = ws;                          // xw (layer1), then xw (layer2)
    float* bufB  = bufA + (size_t)N * D;        // aggregated h (layer1)
    float* bufC  = bufB + (size_t)N * D;        // aggregated z (layer2)
    float* a_src = bufC + (size_t)N * D;        // N*H1
    float* a_dst = a_src + (size_t)N * H1;
    float* amax  = a_dst + (size_t)N * H1;
    float* denom = amax  + (size_t)N * H1;

    const long long ND = (long long)N * D;
    const int blkElems = (int)((ND + 255) / 256);

    // ================= layer 1 (H=4, C=64, concat) =================
    {
        dim3 g(D / (16 * GEMM_WAVES), N / 16);
        gemm_wmma_f32<<<g, 32 * GEMM_WAVES, 0, stream>>>(x, W1, bufA, N, Fin, D);

        const int waves1 = N * H1;
        node_scores<<<(waves1 * 32 + 255) / 256, 256, 0, stream>>>(
            bufA, as1, ad1, a_src, a_dst, amax, denom, N, H1, C1);

        const long long tot1 = (long long)ET * H1;
        edge_softmax_max<<<(int)((tot1 + 255) / 256), 256, 0, stream>>>(
            ei, E, N, a_src, a_dst, amax, H1);
        edge_softmax_sum<<<(int)((tot1 + 255) / 256), 256, 0, stream>>>(
            ei, E, N, a_src, a_dst, amax, denom, H1);

        zero_f32<<<blkElems, 256, 0, stream>>>(bufB, ND);
        edge_aggregate<<<ET, H1 * C1, 0, stream>>>(
            ei, E, N, bufA, a_src, a_dst, amax, denom, bufB, H1, C1);

        bias_act<<<blkElems, 256, 0, stream>>>(bufB, b1, ND, D, /*relu=*/1);
    }

    // ================= layer 2 (H=1, C=256, mean over 1 head == identity) ===
    {
        dim3 g(D / (16 * GEMM_WAVES), N / 16);
        gemm_wmma_f32<<<g, 32 * GEMM_WAVES, 0, stream>>>(bufB, W2, bufA, N, D, D);

        node_scores<<<(N * 32 + 255) / 256, 256, 0, stream>>>(
            bufA, as2, ad2, a_src, a_dst, amax, denom, N, 1, D);

        edge_softmax_max<<<(ET + 255) / 256, 256, 0, stream>>>(
            ei, E, N, a_src, a_dst, amax, 1);
        edge_softmax_sum<<<(ET + 255) / 256, 256, 0, stream>>>(
            ei, E, N, a_src, a_dst, amax, denom, 1);

        zero_f32<<<blkElems, 256, 0, stream>>>(bufC, ND);
        edge_aggregate<<<ET, D, 0, stream>>>(
            ei, E, N, bufA, a_src, a_dst, amax, denom, bufC, 1, D);

        bias_act<<<blkElems, 256, 0, stream>>>(bufC, b2, ND, D, /*relu=*/0);
    }

    // ================= link decode =================
    decode_dot<<<(EL * 32 + 255) / 256, 256, 0, stream>>>(bufC, eli, EL, D, logits);
}